// Cross_PCC_78606491451585
// MI455X (gfx1250) — compile-verified
//
#include <hip/hip_runtime.h>

typedef float v2f __attribute__((ext_vector_type(2)));
typedef float v8f __attribute__((ext_vector_type(8)));

#define IMGF    224.0f
#define INV_IMG (1.0f / 224.0f)
#define Bb 4
#define Vv 8
#define Nn 4096
#define Mm 1024
#define BV (Bb * Vv)

// -------------------------------------------------------------------------
// Kernel A: out_mask[b,v,n] = (padded_mask[xi, yi] == 0)
// -------------------------------------------------------------------------
__global__ void mask_kernel(const float* __restrict__ mask,
                            const float* __restrict__ proj_fine,
                            float* __restrict__ out_mask) {
    int i = blockIdx.x * blockDim.x + threadIdx.x;
    if (i >= BV * Nn) return;
    int bv = i / Nn;
    float pf0 = proj_fine[2 * i];
    float pf1 = proj_fine[2 * i + 1];
    float px = rintf(pf0);
    float py = rintf(IMGF - pf1);
    int xi = (int)py + 1; xi = min(max(xi, 0), 225);
    int yi = (int)px + 1; yi = min(max(yi, 0), 225);
    float mval = 0.0f;
    if (xi >= 1 && xi <= 224 && yi >= 1 && yi <= 224)
        mval = mask[bv * 224 * 224 + (xi - 1) * 224 + (yi - 1)];
    out_mask[i] = (mval == 0.0f) ? 1.0f : 0.0f;
}

// -------------------------------------------------------------------------
// Branchless min/argmin exchange step over a ds_swizzle XOR pattern.
// PAT = (xor_mask << 10) | 0x1F  (group-of-32 mode, and_mask = 0x1F).
// -------------------------------------------------------------------------
template <int PAT>
__device__ __forceinline__ void red_step(float& bval, int& bidx) {
    float ob = __int_as_float(
        __builtin_amdgcn_ds_swizzle(__float_as_int(bval), PAT));
    int oi = __builtin_amdgcn_ds_swizzle(bidx, PAT);
    int take = (int)(ob < bval) | ((int)(ob == bval) & (int)(oi < bidx));
    bval = take ? ob : bval;
    bidx = take ? oi : bidx;
}

// -------------------------------------------------------------------------
// Kernel B: WMMA nearest-neighbor search.
// A block (8 waves) shares one (b,v): its 8KB bounds slab is staged into LDS
// once via GLOBAL_LOAD_ASYNC_TO_LDS_B128 (ASYNCcnt), then each wave runs 64
// V_WMMA_F32_16X16X4_F32 tiles:
//   D[r][c] = |q_c|^2 - 2 p_r . q_c   (coords pre-scaled by 1/224)
//   A row r = (-2 ax_r, -2 ay_r, 1, 0);  B col c = (qx_c, qy_c, |q_c|^2, 0)
// with a running per-lane min/argmin, finished by a ds_swizzle butterfly
// within each 16-lane half (rows 0-7 -> lanes 0-15, rows 8-15 -> 16-31).
// -------------------------------------------------------------------------
__global__ void __launch_bounds__(256)
nn_kernel(const float* __restrict__ bounds,
          const float* __restrict__ proj_fine,
          float* __restrict__ minpart,
          int*   __restrict__ nnidx) {
    __shared__ float sbnd[Mm * 2];          // 8 KB: (qx,qy) pairs for this bv

    int tid  = threadIdx.x;
    int wid  = tid >> 5;
    int lane = tid & 31;
    int bv   = blockIdx.x >> 5;             // 32 blocks per (b,v)
    int tn   = ((blockIdx.x & 31) << 3) + wid;
    int n0   = tn * 16;
    int col  = lane & 15;
    bool hi  = lane >= 16;

    // ---- async-stage bounds[bv] into LDS (256 threads x 2 x 16B = 8KB) ----
    {
        unsigned lds_base = (unsigned)(size_t)&sbnd[0];
        const char* gbase = (const char*)(bounds + (size_t)bv * Mm * 2);
#pragma unroll
        for (int k = 0; k < 2; ++k) {
            unsigned off = (unsigned)((k * 256 + tid) * 16);
            asm volatile("global_load_async_to_lds_b128 %0, %1, %2"
                         :: "v"(lds_base + off), "v"(off), "s"(gbase)
                         : "memory");
        }
        asm volatile("s_wait_asynccnt 0x0" ::: "memory");
    }
    __syncthreads();

    // ---- A matrix (constant across the m loop) ----
    const float* pf = proj_fine + (size_t)(bv * Nn + n0 + col) * 2;
    float ax = rintf(pf[0]) * INV_IMG;
    float ay = rintf(IMGF - pf[1]) * INV_IMG;
    v2f a;
    a.x = hi ? 1.0f : (-2.0f * ax);
    a.y = hi ? 0.0f : (-2.0f * ay);

    v8f czero = {0.0f, 0.0f, 0.0f, 0.0f, 0.0f, 0.0f, 0.0f, 0.0f};

    float best[8];
    int   bidx[8];
#pragma unroll
    for (int j = 0; j < 8; ++j) { best[j] = 3.0e38f; bidx[j] = 0; }

    const float2* sb2 = (const float2*)sbnd;
#pragma unroll 2
    for (int t = 0; t < Mm / 16; ++t) {
        int m = t * 16 + col;
        float2 q = sb2[m];                  // ds_load_b64
        float qx = q.x * INV_IMG;
        float qy = q.y * INV_IMG;
        v2f bb;
        bb.x = hi ? (qx * qx + qy * qy) : qx;
        bb.y = hi ? 0.0f : qy;

        // D = A x B + 0 : v_wmma_f32_16x16x4_f32
        v8f d = __builtin_amdgcn_wmma_f32_16x16x4_f32(
            false, a, false, bb, (short)0, czero, false, false);

#pragma unroll
        for (int j = 0; j < 8; ++j) {
            bool lt = d[j] < best[j];
            best[j] = lt ? d[j] : best[j];
            bidx[j] = lt ? m    : bidx[j];
        }
    }

    // ---- cross-lane min/argmin within each 16-lane half (ds_swizzle) ----
#pragma unroll
    for (int j = 0; j < 8; ++j) {
        red_step<0x201F>(best[j], bidx[j]);   // xor 8
        red_step<0x101F>(best[j], bidx[j]);   // xor 4
        red_step<0x081F>(best[j], bidx[j]);   // xor 2
        red_step<0x041F>(best[j], bidx[j]);   // xor 1
    }

    int base = bv * Nn + n0;
    if (lane == 0) {                 // rows 0..7
#pragma unroll
        for (int j = 0; j < 8; ++j) { minpart[base + j] = best[j]; nnidx[base + j] = bidx[j]; }
    } else if (lane == 16) {         // rows 8..15
#pragma unroll
        for (int j = 0; j < 8; ++j) { minpart[base + 8 + j] = best[j]; nnidx[base + 8 + j] = bidx[j]; }
    }
}

// -------------------------------------------------------------------------
// Kernel C1: per-(b,v) masked mean of dmin (dmin = minpart + |p|^2).
// -------------------------------------------------------------------------
__global__ void dist_kernel(const float* __restrict__ out_mask,
                            const float* __restrict__ minpart,
                            const float* __restrict__ proj_fine,
                            float* __restrict__ dres) {
    int bv  = blockIdx.x;
    int tid = threadIdx.x;
    float s = 0.0f, c = 0.0f;
    for (int n = tid; n < Nn; n += blockDim.x) {
        int i = bv * Nn + n;
        float om  = out_mask[i];
        float pf0 = proj_fine[2 * i];
        float pf1 = proj_fine[2 * i + 1];
        float axp = rintf(pf0) * INV_IMG;
        float ayp = rintf(IMGF - pf1) * INV_IMG;
        float d2  = minpart[i] + axp * axp + ayp * ayp;
        s += om * d2;
        c += om;
    }
    __shared__ float ssum[256];
    __shared__ float scnt[256];
    ssum[tid] = s; scnt[tid] = c;
    __syncthreads();
    for (int off = 128; off > 0; off >>= 1) {
        if (tid < off) { ssum[tid] += ssum[tid + off]; scnt[tid] += scnt[tid + off]; }
        __syncthreads();
    }
    if (tid == 0)
        dres[bv] = (scnt[0] > 0.0f) ? (ssum[0] / fmaxf(scnt[0], 1.0f)) : 0.0f;
}

// -------------------------------------------------------------------------
// Kernel C2: back-project through inv_param for view_id, blend with pc.
// -------------------------------------------------------------------------
__global__ void backproj_kernel(const float* __restrict__ pc,
                                const float* __restrict__ bounds,
                                const int*   __restrict__ view_id,
                                const float* __restrict__ inv_param,
                                const float* __restrict__ proj_finez,
                                const float* __restrict__ out_mask,
                                const int*   __restrict__ nnidx,
                                float* __restrict__ pc_out) {
    int i = blockIdx.x * blockDim.x + threadIdx.x;
    if (i >= Bb * Nn) return;
    int b = i / Nn;
    int n = i % Nn;
    int view = view_id[0];
    int bvn  = (b * Vv + view) * Nn + n;
    float om = out_mask[bvn];
    float o0, o1, o2;
    if (om != 0.0f) {
        int idx = nnidx[bvn];
        const float* bp = bounds + (size_t)((b * Vv + view) * Mm + idx) * 2;
        float z  = proj_finez[bvn];
        float h0 = bp[0] * z;
        float h1 = bp[1] * z;
        float h2 = z;
        float h3 = 1.0f;
        const float* ip = inv_param + (size_t)(b * Vv + view) * 16;
        o0 = h0 * ip[0] + h1 * ip[4] + h2 * ip[8]  + h3 * ip[12];
        o1 = h0 * ip[1] + h1 * ip[5] + h2 * ip[9]  + h3 * ip[13];
        o2 = h0 * ip[2] + h1 * ip[6] + h2 * ip[10] + h3 * ip[14];
    } else {
        o0 = pc[(size_t)i * 3 + 0];
        o1 = pc[(size_t)i * 3 + 1];
        o2 = pc[(size_t)i * 3 + 2];
    }
    pc_out[(size_t)i * 3 + 0] = o0;
    pc_out[(size_t)i * 3 + 1] = o1;
    pc_out[(size_t)i * 3 + 2] = o2;
}

// -------------------------------------------------------------------------
extern "C" void kernel_launch(void* const* d_in, const int* in_sizes, int n_in,
                              void* d_out, int out_size, void* d_ws, size_t ws_size,
                              hipStream_t stream) {
    const float* pc         = (const float*)d_in[0];
    const float* mask       = (const float*)d_in[1];
    const float* bounds     = (const float*)d_in[2];
    const int*   view_id    = (const int*)  d_in[3];
    const float* inv_param  = (const float*)d_in[4];
    const float* proj_fine  = (const float*)d_in[5];
    const float* proj_finez = (const float*)d_in[6];

    float* out    = (float*)d_out;
    float* pc_out = out;                 // (4, 4096, 3)
    float* dres   = out + Bb * Nn * 3;   // (4, 8)

    float* ws       = (float*)d_ws;
    float* out_mask = ws;                       // BV*N floats
    float* minpart  = ws + BV * Nn;             // BV*N floats
    int*   nnidx    = (int*)(ws + 2 * BV * Nn); // BV*N ints

    // A: out_mask
    mask_kernel<<<(BV * Nn + 255) / 256, 256, 0, stream>>>(mask, proj_fine, out_mask);

    // B: WMMA NN search — 32 (b,v) pairs * 256 n-tiles = 8192 waves = 1024 blocks
    int nn_threads = BV * (Nn / 16) * 32;
    nn_kernel<<<nn_threads / 256, 256, 0, stream>>>(bounds, proj_fine, minpart, nnidx);

    // C1: distance_res (one block per (b,v))
    dist_kernel<<<BV, 256, 0, stream>>>(out_mask, minpart, proj_fine, dres);

    // C2: pc_out
    backproj_kernel<<<(Bb * Nn + 255) / 256, 256, 0, stream>>>(
        pc, bounds, view_id, inv_param, proj_finez, out_mask, nnidx, pc_out);
}